// myCTCloss_15075335209007
// MI455X (gfx1250) — compile-verified
//
#include <hip/hip_runtime.h>

// CTC loss forward+backward for B=64, T=768, A=96, L=192 on gfx1250 (MI455X).
// Phase 1/2: latency-bound alpha/beta scans, one WGP per batch element,
//            LDS double-buffer, one barrier per timestep, fast transcendentals.
// Phase 3:   occ = gamma @ onehot via V_WMMA_F32_16X16X4_F32 (exact fp32 WMMA),
//            fused with grads = outputs - occ and the (b,t,a)->(b,a,t) transpose.

#define NEGV (-1e30f)

constexpr int B_  = 64;
constexpr int T_  = 768;
constexpr int A_  = 96;
constexpr int L_  = 192;
constexpr int S_  = 2 * L_ + 1;   // 385
constexpr int SP_ = 416;          // S padded to 13 * 32 for the WMMA K loop

typedef __attribute__((ext_vector_type(2))) float v2f;
typedef __attribute__((ext_vector_type(8))) float v8f;

__device__ __forceinline__ float lse3f(float a, float b, float c) {
    float m = fmaxf(a, fmaxf(b, c));
    return m + __logf(__expf(a - m) + __expf(b - m) + __expf(c - m));
}
__device__ __forceinline__ float lpclamp(float p) {
    return __logf(fminf(fmaxf(p, 1e-30f), 1.0f));
}

// ---------------------------------------------------------------------------
// Forward alpha scan: one block (13 wave32s) per batch element.
// la kept in LDS, double-buffered -> one s_barrier per timestep.
// ---------------------------------------------------------------------------
__global__ __launch_bounds__(SP_) void ctc_forward(
    const float* __restrict__ outputs, const int* __restrict__ labels,
    const int* __restrict__ tlens, float* __restrict__ la_all,
    float* __restrict__ logZ, float* __restrict__ nll)
{
    __shared__ float bufA[S_ + 2];
    __shared__ float bufB[S_ + 2];
    const int b = blockIdx.x;
    const int s = threadIdx.x;
    const int tl = tlens[b];
    const int Smax = 2 * tl + 1;

    int  ext_s = 0;
    bool allow = false;
    if (s < S_ && (s & 1)) {
        int li = (s - 1) >> 1;
        ext_s = labels[b * L_ + li];
        allow = (li == 0) || (ext_s != labels[b * L_ + li - 1]);
    }
    const bool vs = (s < S_) && (s < Smax);
    const float* op  = outputs + (size_t)b * T_ * A_;
    float*       lab = la_all  + (size_t)b * T_ * S_;

    if (s < 2) { bufA[s] = NEGV; bufB[s] = NEGV; }   // left padding (never rewritten)
    float* cur = bufA;
    float* nxt = bufB;

    // t = 0 : la0[s] = (s < 2) ? lp : NEG
    if (s < S_) {
        float lp = vs ? lpclamp(op[ext_s]) : NEGV;
        float v  = (s < 2) ? lp : NEGV;
        cur[s + 2] = v;
        lab[s] = v;
    }
    __syncthreads();

    for (int t = 1; t < T_; ++t) {
        if (s < S_) {
            if (t + 8 < T_) __builtin_prefetch(op + (t + 8) * A_ + ext_s, 0, 1);
            float lp = vs ? lpclamp(op[t * A_ + ext_s]) : NEGV;
            float a0 = cur[s + 2];
            float a1 = cur[s + 1];
            float a2 = allow ? cur[s] : NEGV;
            float v  = lse3f(a0, a1, a2) + lp;
            nxt[s + 2] = v;
            lab[t * S_ + s] = v;
        }
        __syncthreads();
        float* tmp = cur; cur = nxt; nxt = tmp;
    }

    if (s == 0) {
        float x = cur[2 * tl + 1];   // laT[2*tl - 1]  (+2 pad offset)
        float y = cur[2 * tl + 2];   // laT[2*tl]
        float m = fmaxf(x, y);
        float lz = m + __logf(__expf(x - m) + __expf(y - m));
        logZ[b] = lz;
        nll[b]  = -lz;
    }
}

// ---------------------------------------------------------------------------
// Backward beta scan fused with gamma = exp(la + lb - lp - logZ).
// ---------------------------------------------------------------------------
__global__ __launch_bounds__(SP_) void ctc_backward(
    const float* __restrict__ outputs, const int* __restrict__ labels,
    const int* __restrict__ tlens, const float* __restrict__ la_all,
    const float* __restrict__ logZ, float* __restrict__ gamma)
{
    __shared__ float bufA[S_ + 2];
    __shared__ float bufB[S_ + 2];
    const int b = blockIdx.x;
    const int s = threadIdx.x;
    const int tl = tlens[b];
    const int Smax = 2 * tl + 1;

    int ext_s = 0;
    if (s < S_ && (s & 1)) ext_s = labels[b * L_ + ((s - 1) >> 1)];
    bool allowb = false;
    if (s < S_ - 2 && (s & 1)) {                 // allow[s+2]; s+2 odd, li2 >= 1
        int li2 = (s + 1) >> 1;
        allowb = (labels[b * L_ + li2] != labels[b * L_ + li2 - 1]);
    }
    const bool vs = (s < S_) && (s < Smax);
    const float* op  = outputs + (size_t)b * T_ * A_;
    const float* lab = la_all  + (size_t)b * T_ * S_;
    float*       gm  = gamma   + (size_t)b * T_ * S_;
    const float  lz  = logZ[b];

    if (s < 2) { bufA[S_ + s] = NEGV; bufB[S_ + s] = NEGV; }  // right padding
    float* cur = bufA;
    float* nxt = bufB;

    // t = T-1 : lbT = end_mask ? lp : NEG
    if (s < S_) {
        float lp  = vs ? lpclamp(op[(T_ - 1) * A_ + ext_s]) : NEGV;
        bool endm = (s == 2 * tl) || (s == 2 * tl - 1);
        float v   = endm ? lp : NEGV;
        cur[s] = v;
        gm[(T_ - 1) * S_ + s] = vs ? __expf(lab[(T_ - 1) * S_ + s] + v - lp - lz) : 0.0f;
    }
    __syncthreads();

    for (int t = T_ - 2; t >= 0; --t) {
        if (s < S_) {
            if (t - 8 >= 0) __builtin_prefetch(op + (t - 8) * A_ + ext_s, 0, 1);
            float lp = vs ? lpclamp(op[t * A_ + ext_s]) : NEGV;
            float b0 = cur[s];
            float b1 = cur[s + 1];
            float b2 = allowb ? cur[s + 2] : NEGV;
            float v  = lse3f(b0, b1, b2) + lp;
            nxt[s] = v;
            gm[t * S_ + s] = vs ? __expf(lab[t * S_ + s] + v - lp - lz) : 0.0f;
        }
        __syncthreads();
        float* tmp = cur; cur = nxt; nxt = tmp;
    }
}

// ---------------------------------------------------------------------------
// Build dense fp32 one-hot of extended labels, zero-padded to SP_ rows.
// ---------------------------------------------------------------------------
__global__ void build_onehot(const int* __restrict__ labels,
                             float* __restrict__ onehot, int total)
{
    int idx = blockIdx.x * blockDim.x + threadIdx.x;
    if (idx >= total) return;
    int a    = idx % A_;
    int rest = idx / A_;
    int s    = rest % SP_;
    int b    = rest / SP_;
    float v = 0.0f;
    if (s < S_) {
        int e = (s & 1) ? labels[b * L_ + ((s - 1) >> 1)] : 0;
        v = (e == a) ? 1.0f : 0.0f;
    }
    onehot[idx] = v;
}

// ---------------------------------------------------------------------------
// occ = gamma (T x S) @ onehot (S x A) per batch, via V_WMMA_F32_16X16X4_F32.
// Block: 256 threads = 8 waves; block tile 64(M=t) x 32(N=a); K tiled by 32.
// Fused epilogue: grads[b,a,t] = outputs[b,t,a] - occ[t,a].
// ---------------------------------------------------------------------------
__global__ __launch_bounds__(256) void occ_grads_wmma(
    const float* __restrict__ gamma, const float* __restrict__ onehot,
    const float* __restrict__ outputs, float* __restrict__ grads)
{
    __shared__ float gA[64][33];   // gamma tile  [m][k], pad avoids bank conflicts
    __shared__ float gB[32][33];   // onehot tile [k][n]
    const int b    = blockIdx.x;
    const int m0   = blockIdx.y * 64;
    const int n0   = blockIdx.z * 32;
    const int tid  = threadIdx.x;
    const int lane = tid & 31;
    const int wv   = tid >> 5;            // 0..7
    const int mt   = (wv & 3) * 16;       // wave M tile within block
    const int nt   = (wv >> 2) * 16;      // wave N tile within block
    const float* gmb = gamma  + (size_t)b * T_ * S_;
    const float* ohb = onehot + (size_t)b * SP_ * A_;

    const int hi = lane >> 4;             // half-wave select
    const int mr = mt + (lane & 15);
    const int nc = nt + (lane & 15);

    v8f acc = {};
    for (int k0 = 0; k0 < SP_; k0 += 32) {
        // Stage gamma 64x32 (coalesced 128B rows; zero-fill k >= S_)
        for (int i = tid; i < 64 * 32; i += 256) {
            int r = i >> 5, k = i & 31;
            int kk = k0 + k;
            gA[r][k] = (kk < S_) ? gmb[(m0 + r) * S_ + kk] : 0.0f;
        }
        // Stage onehot 32x32 (already zero-padded rows for k >= S_)
        for (int i = tid; i < 32 * 32; i += 256) {
            int r = i >> 5, c = i & 31;
            gB[r][c] = ohb[(k0 + r) * A_ + (n0 + c)];
        }
        __syncthreads();
        #pragma unroll
        for (int kk = 0; kk < 32; kk += 4) {
            // ISA 7.12.2 layouts:
            // A 16x4 : lanes 0-15 K={0,1}, lanes 16-31 K={2,3}
            // B 4x16 : VGPR0 K={0,1} by half-wave, VGPR1 K={2,3}
            v2f av, bv;
            av.x = gA[mr][kk + 2 * hi];
            av.y = gA[mr][kk + 2 * hi + 1];
            bv.x = gB[kk + hi][nc];
            bv.y = gB[kk + 2 + hi][nc];
            acc = __builtin_amdgcn_wmma_f32_16x16x4_f32(
                false, av, false, bv, (short)0, acc, false, false);
        }
        __syncthreads();
    }

    // Epilogue: grads[b, a, t] = outputs[b, t, a] - occ[t, a]
    const float* opb = outputs + (size_t)b * T_ * A_;
    float*       gr  = grads   + (size_t)b * A_ * T_;
    const int ag = n0 + nt + (lane & 15);
    #pragma unroll
    for (int i = 0; i < 8; ++i) {
        int tg = m0 + mt + i + (hi << 3);   // D layout: VGPR i -> M = i (+8 hi)
        float g = opb[tg * A_ + ag] - acc[i];
        gr[(size_t)ag * T_ + tg] = g;
    }
}

// ---------------------------------------------------------------------------
// loss = mean(nll); tiny deterministic reduction (B_ = 64).
// ---------------------------------------------------------------------------
__global__ void finish_loss(const float* __restrict__ nll, float* __restrict__ out)
{
    if (threadIdx.x == 0 && blockIdx.x == 0) {
        float acc = 0.0f;
        for (int i = 0; i < B_; ++i) acc += nll[i];
        out[0] = acc / (float)B_;
    }
}

extern "C" void kernel_launch(void* const* d_in, const int* in_sizes, int n_in,
                              void* d_out, int out_size, void* d_ws, size_t ws_size,
                              hipStream_t stream)
{
    const float* outputs = (const float*)d_in[0];   // (B,T,A) f32
    const int*   labels  = (const int*)d_in[1];     // (B,L)   i32
    const int*   tlens   = (const int*)d_in[2];     // (B,)    i32
    float* out = (float*)d_out;                     // [loss(1), grads(B*A*1*T)]

    float* ws = (float*)d_ws;
    const size_t nTS = (size_t)B_ * T_ * S_;
    float* la     = ws;                                   // B*T*S
    float* gamma  = la + nTS;                             // B*T*S
    float* onehot = gamma + nTS;                          // B*SP*A
    float* logZ   = onehot + (size_t)B_ * SP_ * A_;       // B
    float* nll    = logZ + B_;                            // B

    const int ohTotal = B_ * SP_ * A_;
    build_onehot<<<(ohTotal + 255) / 256, 256, 0, stream>>>(labels, onehot, ohTotal);
    ctc_forward<<<B_, SP_, 0, stream>>>(outputs, labels, tlens, la, logZ, nll);
    ctc_backward<<<B_, SP_, 0, stream>>>(outputs, labels, tlens, la, logZ, gamma);
    dim3 grid(B_, T_ / 64, A_ / 32);
    occ_grads_wmma<<<grid, 256, 0, stream>>>(gamma, onehot, outputs, out + 1);
    finish_loss<<<1, 32, 0, stream>>>(nll, out);
}